// HALEAttention_34359738368293
// MI455X (gfx1250) — compile-verified
//
#include <hip/hip_runtime.h>
#include <hip/hip_bf16.h>

typedef float v2f __attribute__((ext_vector_type(2)));
typedef float v8f __attribute__((ext_vector_type(8)));

#define NTOK   2048
#define DMODEL 1024
#define NHEAD  16
#define DHEAD  64
#define CHUNK  128
#define NCHUNK 16
#define WIN    64
#define SZ     (NTOK * DMODEL)   // 2M elements

__device__ __forceinline__ v8f wmma4(v2f a, v2f b, v8f c) {
    // D = A(16x4 f32) * B(4x16 f32) + C(16x16 f32)
    return __builtin_amdgcn_wmma_f32_16x16x4_f32(false, a, false, b, (short)0, c, false, false);
}

__device__ __forceinline__ float phi(float x) {   // elu(x)+1
    return x > 0.f ? x + 1.f : __expf(x);
}

// -------------------------------------------------------------------------
// Generic batched GEMM: C[m,n] = act( sum_k A[m,k]*B[n,k] + bias[n] )
// A row-major (lda), B row-major [N,K] (ldb), C row-major (ldc).
// grid = (N/64, M/16, batch); block = 128 (4 waves, one 16x16 tile each).
// -------------------------------------------------------------------------
__global__ void wmma_gemm_f32(const float* __restrict__ A, const float* __restrict__ B,
                              float* __restrict__ C, const float* __restrict__ bias,
                              int M, int N, int K, int lda, int ldb, int ldc,
                              long sA, long sB, long sC, int act)
{
    const int wave = threadIdx.x >> 5;
    const int lane = threadIdx.x & 31;
    const int half = lane >> 4;       // 0: K+0/1 , 1: K+2/3
    const int l16  = lane & 15;
    const int nBase = (blockIdx.x * 4 + wave) * 16;
    const int mBase = blockIdx.y * 16;
    const int b = blockIdx.z;

    const float* arow = A + (long)b * sA + (long)(mBase + l16) * lda;
    const float* brow = B + (long)b * sB + (long)(nBase + l16) * ldb;

    v8f acc; 
    #pragma unroll
    for (int i = 0; i < 8; ++i) acc[i] = 0.f;

    for (int kk = 0; kk < K; kk += 64) {
        __builtin_prefetch(arow + kk + 64, 0, 0);
        __builtin_prefetch(brow + kk + 64, 0, 0);
        #pragma unroll
        for (int k = 0; k < 64; k += 4) {
            const int k0 = kk + k + half * 2;
            v2f av = *(const v2f*)(arow + k0);
            v2f bv = *(const v2f*)(brow + k0);
            acc = wmma4(av, bv, acc);
        }
    }

    const float bb = bias ? bias[nBase + l16] : 0.f;
    float* crow = C + (long)b * sC + (long)(nBase + l16);
    #pragma unroll
    for (int r = 0; r < 8; ++r) {
        const int m = mBase + r + half * 8;
        float v = acc[r] + bb;
        if (act == 2) v = v / (1.f + __expf(-v));   // silu
        crow[(long)m * ldc] = v;
    }
}

// -------------------------------------------------------------------------
// Local causal window attention (raw q, k_loc, v_loc). One thread per (n,h).
// Layout: [N, H*DHEAD].
// -------------------------------------------------------------------------
__global__ void local_attn_k(const float* __restrict__ Q, const float* __restrict__ KL,
                             const float* __restrict__ VL, float* __restrict__ outm)
{
    const int t = blockIdx.x * blockDim.x + threadIdx.x;
    if (t >= NTOK * NHEAD) return;
    const int h = t & (NHEAD - 1);
    const int n = t >> 4;
    const float scale = 0.125f;   // 1/sqrt(64)
    const float* q = Q + (long)n * DMODEL + h * DHEAD;
    const int j0 = (n - (WIN - 1)) < 0 ? 0 : (n - (WIN - 1));

    float mx = -1e30f;
    for (int j = j0; j <= n; ++j) {
        const float* kr = KL + (long)j * DMODEL + h * DHEAD;
        float s = 0.f;
        #pragma unroll
        for (int d = 0; d < DHEAD; ++d) s += q[d] * kr[d];
        s *= scale;
        mx = fmaxf(mx, s);
    }
    float denom = 0.f;
    float acc[DHEAD];
    #pragma unroll
    for (int d = 0; d < DHEAD; ++d) acc[d] = 0.f;
    for (int j = j0; j <= n; ++j) {
        const float* kr = KL + (long)j * DMODEL + h * DHEAD;
        float s = 0.f;
        #pragma unroll
        for (int d = 0; d < DHEAD; ++d) s += q[d] * kr[d];
        const float p = __expf(s * scale - mx);
        denom += p;
        const float* vr = VL + (long)j * DMODEL + h * DHEAD;
        #pragma unroll
        for (int d = 0; d < DHEAD; ++d) acc[d] += p * vr[d];
    }
    float* o = outm + (long)n * DMODEL + h * DHEAD;
    const float inv = 1.f / denom;
    #pragma unroll
    for (int d = 0; d < DHEAD; ++d) o[d] = acc[d] * inv;
}

// -------------------------------------------------------------------------
// Per-chunk sums: kvsum[h,c,d,e] = sum_j phi(k[j,d])*v[j,e]; ksum[h,c,d].
// grid = H*NC blocks, 256 threads, 64KB dynamic LDS.
// -------------------------------------------------------------------------
__global__ void chunk_sums_k(const float* __restrict__ Karr, const float* __restrict__ Varr,
                             float* __restrict__ kvsums, float* __restrict__ ksums)
{
    const int h = blockIdx.x & (NHEAD - 1);
    const int c = blockIdx.x >> 4;
    extern __shared__ float sm[];
    float* kc = sm;                 // 128*64
    float* vc = sm + CHUNK * DHEAD;

    for (int i = threadIdx.x; i < CHUNK * DHEAD; i += blockDim.x) {
        const int j = i >> 6, d = i & 63;
        const long g = (long)(c * CHUNK + j) * DMODEL + h * DHEAD + d;
        kc[i] = phi(Karr[g]);
        vc[i] = Varr[g];
    }
    __syncthreads();

    const int e  = threadIdx.x & 63;
    const int dB = (threadIdx.x >> 6) * 16;
    float acc[16];
    #pragma unroll
    for (int i = 0; i < 16; ++i) acc[i] = 0.f;
    for (int j = 0; j < CHUNK; ++j) {
        const float vv = vc[j * DHEAD + e];
        #pragma unroll
        for (int dd = 0; dd < 16; ++dd) acc[dd] += kc[j * DHEAD + dB + dd] * vv;
    }
    float* out = kvsums + (long)(h * NCHUNK + c) * (DHEAD * DHEAD);
    #pragma unroll
    for (int dd = 0; dd < 16; ++dd) out[(dB + dd) * DHEAD + e] = acc[dd];

    if (threadIdx.x < DHEAD) {
        float s = 0.f;
        for (int j = 0; j < CHUNK; ++j) s += kc[j * DHEAD + threadIdx.x];
        ksums[(h * NCHUNK + c) * DHEAD + threadIdx.x] = s;
    }
}

// Exclusive prefix over chunks. grid = H blocks.
__global__ void chunk_prefix_k(const float* __restrict__ kvsums, const float* __restrict__ ksums,
                               float* __restrict__ kvpref, float* __restrict__ skpref)
{
    const int h = blockIdx.x;
    for (int idx = threadIdx.x; idx < DHEAD * DHEAD; idx += blockDim.x) {
        float run = 0.f;
        for (int c = 0; c < NCHUNK; ++c) {
            const long o = (long)(h * NCHUNK + c) * (DHEAD * DHEAD) + idx;
            kvpref[o] = run;
            run += kvsums[o];
        }
    }
    if (threadIdx.x < DHEAD) {
        float run = 0.f;
        for (int c = 0; c < NCHUNK; ++c) {
            const int o = (h * NCHUNK + c) * DHEAD + threadIdx.x;
            skpref[o] = run;
            run += ksums[o];
        }
    }
}

// -------------------------------------------------------------------------
// Chunked causal linear attention body (WMMA). grid = H*NC, 256 thr (8 waves).
// lvl < 0 : globm = out/norm ; lvl >= 0 : globm += softmax(hs)[lvl]*out/norm.
// -------------------------------------------------------------------------
__global__ void lin_attn_k(const float* __restrict__ Qarr, const float* __restrict__ Karr,
                           const float* __restrict__ Varr, const float* __restrict__ kvpref,
                           const float* __restrict__ skpref, float* __restrict__ globm,
                           const float* __restrict__ hs, int lvl)
{
    const int h = blockIdx.x & (NHEAD - 1);
    const int c = blockIdx.x >> 4;
    extern __shared__ float sm[];
    float* qc  = sm;                      // 128*64 phi(q)
    float* kc  = qc + CHUNK * DHEAD;      // 128*64 phi(k)
    float* vc  = kc + CHUNK * DHEAD;      // 128*64 v
    float* Ob  = vc + CHUNK * DHEAD;      // 128*64 output
    float* nrm = Ob + CHUNK * DHEAD;      // 128 row norms (intra part)
    float* stg = nrm + CHUNK;             // 8 waves * 16*16 staging

    const int wave = threadIdx.x >> 5;
    const int lane = threadIdx.x & 31;
    const int half = lane >> 4;
    const int l16  = lane & 15;
    float* st = stg + wave * 256;

    for (int i = threadIdx.x; i < CHUNK * DHEAD; i += blockDim.x) {
        const int j = i >> 6, d = i & 63;
        const long g = (long)(c * CHUNK + j) * DMODEL + h * DHEAD + d;
        qc[i] = phi(Qarr[g]);
        kc[i] = phi(Karr[g]);
        vc[i] = Varr[g];
    }
    if (threadIdx.x < CHUNK) nrm[threadIdx.x] = 0.f;
    __syncthreads();

    const int mi = wave;                              // 8 m-tiles of 16 rows
    const float* arow = qc + (mi * 16 + l16) * DHEAD; // phi(q) fragment row

    v8f o[4];
    #pragma unroll
    for (int nt = 0; nt < 4; ++nt)
        #pragma unroll
        for (int r = 0; r < 8; ++r) o[nt][r] = 0.f;
    float rs[8];
    #pragma unroll
    for (int r = 0; r < 8; ++r) rs[r] = 0.f;

    // inter-chunk: O += phi(q) @ S_kv_prefix  (K = 64)
    const float* kvp = kvpref + (long)(h * NCHUNK + c) * (DHEAD * DHEAD);
    for (int k = 0; k < DHEAD; k += 4) {
        const int k0 = k + half * 2;
        v2f av = *(const v2f*)(arow + k0);
        #pragma unroll
        for (int nt = 0; nt < 4; ++nt) {
            v2f bv;
            bv.x = kvp[k0 * DHEAD + nt * 16 + l16];
            bv.y = kvp[(k0 + 1) * DHEAD + nt * 16 + l16];
            o[nt] = wmma4(av, bv, o[nt]);
        }
    }

    // intra-chunk causal: S = phi(q) phi(k)^T, masked; O += S @ v
    for (int jt = 0; jt <= mi; ++jt) {
        v8f s;
        #pragma unroll
        for (int r = 0; r < 8; ++r) s[r] = 0.f;
        const float* brow = kc + (jt * 16 + l16) * DHEAD;
        for (int k = 0; k < DHEAD; k += 4) {
            const int k0 = k + half * 2;
            v2f av = *(const v2f*)(arow + k0);
            v2f bv = *(const v2f*)(brow + k0);
            s = wmma4(av, bv, s);
        }
        // mask (causal, inclusive diagonal), row-sum partials, stage to LDS
        #pragma unroll
        for (int r = 0; r < 8; ++r) {
            const int ml = r + half * 8;            // row within tile
            float v = s[r];
            v = (jt == mi && l16 > ml) ? 0.f : v;
            rs[r] += v;
            st[ml * 16 + l16] = v;                  // C-layout -> row-major
        }
        asm volatile("s_wait_dscnt 0" ::: "memory");
        // O += S_tile(16x16) @ v_tile(16x64)
        #pragma unroll
        for (int k = 0; k < 16; k += 4) {
            const int k0 = k + half * 2;
            v2f av = *(const v2f*)(st + l16 * 16 + k0);
            #pragma unroll
            for (int nt = 0; nt < 4; ++nt) {
                v2f bv;
                bv.x = vc[(jt * 16 + k0) * DHEAD + nt * 16 + l16];
                bv.y = vc[(jt * 16 + k0 + 1) * DHEAD + nt * 16 + l16];
                o[nt] = wmma4(av, bv, o[nt]);
            }
        }
        asm volatile("s_wait_dscnt 0" ::: "memory");
    }

    // write O tiles + accumulate row norms
    #pragma unroll
    for (int r = 0; r < 8; ++r) {
        const int m = mi * 16 + r + half * 8;
        #pragma unroll
        for (int nt = 0; nt < 4; ++nt) Ob[m * DHEAD + nt * 16 + l16] = o[nt][r];
        atomicAdd(&nrm[m], rs[r]);                  // ds_add_f32
    }
    __syncthreads();

    // normalize + (optionally weighted) writeout
    float w = 1.f;
    if (lvl >= 0) {
        const float h0 = hs[0], h1 = hs[1], h2 = hs[2], h3 = hs[3];
        const float m4 = fmaxf(fmaxf(h0, h1), fmaxf(h2, h3));
        const float e0 = __expf(h0 - m4), e1 = __expf(h1 - m4),
                    e2 = __expf(h2 - m4), e3 = __expf(h3 - m4);
        const float es[4] = {e0, e1, e2, e3};
        w = es[lvl] / (e0 + e1 + e2 + e3);
    }
    const float* skp = skpref + (h * NCHUNK + c) * DHEAD;
    for (int row = threadIdx.x; row < CHUNK; row += blockDim.x) {
        float nv = nrm[row];
        const float* qr = qc + row * DHEAD;
        #pragma unroll
        for (int d = 0; d < DHEAD; ++d) nv += qr[d] * skp[d];
        const float inv = 1.f / fmaxf(nv, 1e-6f);
        const long gbase = (long)(c * CHUNK + row) * DMODEL + h * DHEAD;
        if (lvl < 0) {
            #pragma unroll
            for (int d = 0; d < DHEAD; ++d) globm[gbase + d] = Ob[row * DHEAD + d] * inv;
        } else {
            #pragma unroll
            for (int d = 0; d < DHEAD; ++d) globm[gbase + d] += w * Ob[row * DHEAD + d] * inv;
        }
    }
}

// causal block means of k,v (raw). thread per (n, col).
__global__ void block_mean_k(const float* __restrict__ Karr, const float* __restrict__ Varr,
                             float* __restrict__ bmk, float* __restrict__ bmv, int block)
{
    const long t = (long)blockIdx.x * blockDim.x + threadIdx.x;
    if (t >= (long)SZ) return;
    const int col = (int)(t & (DMODEL - 1));
    const int n   = (int)(t >> 10);
    const int start = (n / block) * block;
    float sk = 0.f, sv = 0.f;
    for (int j = start; j <= n; ++j) {
        sk += Karr[(long)j * DMODEL + col];
        sv += Varr[(long)j * DMODEL + col];
    }
    const float inv = 1.f / (float)(n - start + 1);
    bmk[t] = sk * inv;
    bmv[t] = sv * inv;
}

// cat = [x, local_m - glob_m]
__global__ void cat_k(const float* __restrict__ x, const float* __restrict__ localm,
                      const float* __restrict__ globm, float* __restrict__ cat)
{
    const long t = (long)blockIdx.x * blockDim.x + threadIdx.x;
    if (t >= (long)NTOK * 2 * DMODEL) return;
    const int col = (int)(t & (2 * DMODEL - 1));
    const long n  = t >> 11;
    cat[t] = (col < DMODEL) ? x[n * DMODEL + col]
                            : (localm[n * DMODEL + col - DMODEL] - globm[n * DMODEL + col - DMODEL]);
}

// alpha[n] = sigmoid(gate_h[n] . Wgo + bgo)
__global__ void alpha_k(const float* __restrict__ gateh, const float* __restrict__ Wgo,
                        const float* __restrict__ bgo, float* __restrict__ alpha)
{
    const int n = blockIdx.x * blockDim.x + threadIdx.x;
    if (n >= NTOK) return;
    float s = bgo[0];
    for (int j = 0; j < DMODEL; ++j) s += gateh[(long)n * DMODEL + j] * Wgo[j];
    alpha[n] = 1.f / (1.f + __expf(-s));
}

// mixed = a*local + (1-a)*glob
__global__ void mixed_k(const float* __restrict__ alpha, const float* __restrict__ localm,
                        const float* __restrict__ globm, float* __restrict__ mixed)
{
    const long t = (long)blockIdx.x * blockDim.x + threadIdx.x;
    if (t >= (long)SZ) return;
    const float a = alpha[t >> 10];
    mixed[t] = a * localm[t] + (1.f - a) * globm[t];
}

// -------------------------------------------------------------------------
extern "C" void kernel_launch(void* const* d_in, const int* in_sizes, int n_in,
                              void* d_out, int out_size, void* d_ws, size_t ws_size,
                              hipStream_t stream)
{
    (void)in_sizes; (void)n_in; (void)out_size; (void)ws_size;
    const float* x       = (const float*)d_in[0];
    const float* Wq      = (const float*)d_in[1];
    const float* Wk      = (const float*)d_in[2];
    const float* Wv      = (const float*)d_in[3];
    const float* Wkl     = (const float*)d_in[4];
    const float* Wvl     = (const float*)d_in[5];
    const float* haar_Wk = (const float*)d_in[6];
    const float* haar_Wv = (const float*)d_in[7];
    const float* haar_s  = (const float*)d_in[8];
    const float* Wg      = (const float*)d_in[9];
    const float* bg      = (const float*)d_in[10];
    const float* Wgo     = (const float*)d_in[11];
    const float* bgo     = (const float*)d_in[12];
    const float* Wo      = (const float*)d_in[13];
    const float* bo      = (const float*)d_in[14];
    float* out = (float*)d_out;

    float* ws = (float*)d_ws;
    float* Q      = ws;
    float* Kp     = ws + 1L * SZ;
    float* V      = ws + 2L * SZ;
    float* KL     = ws + 3L * SZ;
    float* VL     = ws + 4L * SZ;
    float* localm = ws + 5L * SZ;
    float* globm  = ws + 6L * SZ;
    float* kvsums = ws + 7L * SZ;                   // SZ/2
    float* kvpref = ws + 7L * SZ + SZ / 2;          // SZ/2
    float* ksums  = ws + 8L * SZ;                   // 16K
    float* skpref = ws + 8L * SZ + 16384;           // 16K
    float* alpha  = ws + 8L * SZ + 32768;           // 2K
    float* bmk    = ws + 9L * SZ;
    float* bmv    = ws + 10L * SZ;
    float* klvl   = ws + 11L * SZ;
    float* vlvl   = ws + 12L * SZ;
    float* cat    = Q;     // reuse Q+Kp region (4M floats) after attention
    float* gateh  = KL;    // reuse
    float* mixed  = V;     // reuse

    auto gemm = [&](const float* A, const float* B, float* C, const float* bias,
                    int M, int N, int K, int lda, int ldb, int ldc,
                    long sA, long sB, long sC, int batch, int act) {
        dim3 grid(N / 64, M / 16, batch);
        wmma_gemm_f32<<<grid, dim3(128), 0, stream>>>(A, B, C, bias, M, N, K,
                                                      lda, ldb, ldc, sA, sB, sC, act);
    };

    // projections
    gemm(x, Wq,  Q,  nullptr, NTOK, DMODEL, DMODEL, DMODEL, DMODEL, DMODEL, 0, 0, 0, 1, 0);
    gemm(x, Wk,  Kp, nullptr, NTOK, DMODEL, DMODEL, DMODEL, DMODEL, DMODEL, 0, 0, 0, 1, 0);
    gemm(x, Wv,  V,  nullptr, NTOK, DMODEL, DMODEL, DMODEL, DMODEL, DMODEL, 0, 0, 0, 1, 0);
    gemm(x, Wkl, KL, nullptr, NTOK, DMODEL, DMODEL, DMODEL, DMODEL, DMODEL, 0, 0, 0, 1, 0);
    gemm(x, Wvl, VL, nullptr, NTOK, DMODEL, DMODEL, DMODEL, DMODEL, DMODEL, 0, 0, 0, 1, 0);

    // local window attention
    local_attn_k<<<dim3((NTOK * NHEAD) / 256), dim3(256), 0, stream>>>(Q, KL, VL, localm);

    const size_t shSums = 2 * CHUNK * DHEAD * sizeof(float);                       // 64 KB
    const size_t shLin  = (4 * CHUNK * DHEAD + CHUNK + 8 * 256) * sizeof(float);   // ~136.5 KB

    // main chunked causal linear attention
    chunk_sums_k<<<dim3(NHEAD * NCHUNK), dim3(256), shSums, stream>>>(Kp, V, kvsums, ksums);
    chunk_prefix_k<<<dim3(NHEAD), dim3(256), 0, stream>>>(kvsums, ksums, kvpref, skpref);
    lin_attn_k<<<dim3(NHEAD * NCHUNK), dim3(256), shLin, stream>>>(Q, Kp, V, kvpref, skpref,
                                                                   globm, haar_s, -1);

    // Haar levels
    for (int lvl = 0; lvl < 4; ++lvl) {
        const int block = 2 << lvl;  // 2,4,8,16
        block_mean_k<<<dim3(SZ / 256), dim3(256), 0, stream>>>(Kp, V, bmk, bmv, block);
        gemm(bmk, haar_Wk + (long)lvl * DHEAD * DHEAD, klvl, nullptr,
             NTOK, DHEAD, DHEAD, DMODEL, DHEAD, DMODEL, 64, 0, 64, NHEAD, 0);
        gemm(bmv, haar_Wv + (long)lvl * DHEAD * DHEAD, vlvl, nullptr,
             NTOK, DHEAD, DHEAD, DMODEL, DHEAD, DMODEL, 64, 0, 64, NHEAD, 0);
        chunk_sums_k<<<dim3(NHEAD * NCHUNK), dim3(256), shSums, stream>>>(klvl, vlvl, kvsums, ksums);
        chunk_prefix_k<<<dim3(NHEAD), dim3(256), 0, stream>>>(kvsums, ksums, kvpref, skpref);
        lin_attn_k<<<dim3(NHEAD * NCHUNK), dim3(256), shLin, stream>>>(Q, klvl, vlvl, kvpref,
                                                                       skpref, globm, haar_s, lvl);
    }

    // gate
    cat_k<<<dim3((2 * SZ) / 256), dim3(256), 0, stream>>>(x, localm, globm, cat);
    gemm(cat, Wg, gateh, bg, NTOK, DMODEL, 2 * DMODEL, 2 * DMODEL, 2 * DMODEL, DMODEL,
         0, 0, 0, 1, 2 /*silu*/);
    alpha_k<<<dim3(NTOK / 256), dim3(256), 0, stream>>>(gateh, Wgo, bgo, alpha);
    mixed_k<<<dim3(SZ / 256), dim3(256), 0, stream>>>(alpha, localm, globm, mixed);

    // output projection
    gemm(mixed, Wo, out, bo, NTOK, DMODEL, DMODEL, DMODEL, DMODEL, DMODEL, 0, 0, 0, 1, 0);
}